// LogMultiDenseExp_84902913507663
// MI455X (gfx1250) — compile-verified
//
#include <hip/hip_runtime.h>
#include <cstdint>

// ---------------------------------------------------------------------------
// GOOM log_matmul_exp for [8,256,128,64] complex64 inputs (d = n = 64).
// One workgroup (256 threads = 8 wave32) per batch element (2048 total).
// Complex GEMM packed as a single real f32 GEMM 128x128x128 done with
// V_WMMA_F32_16X16X4_F32. The [zeros; I] concat block is handled as a cheap
// per-row epilogue correction instead of widening the GEMM.
// Memory-bound (384 MiB @ 23.3 TB/s ~= 16.5us) so f32 WMMA precision is free.
// Packed B is stored TRANSPOSED in LDS so each WMMA B fragment is one
// contiguous ds_load_b64 into an even VGPR pair (no v_mov repacking).
// Output staged in LDS and streamed out via global_store_async_from_lds_b128.
// ---------------------------------------------------------------------------

typedef float v2f __attribute__((ext_vector_type(2)));
typedef float v8f __attribute__((ext_vector_type(8)));
typedef int   v4i __attribute__((ext_vector_type(4)));

#define AS1 __attribute__((address_space(1)))
#define AS3 __attribute__((address_space(3)))

#if defined(__has_builtin)
#if __has_builtin(__builtin_amdgcn_global_store_async_from_lds_b128)
#define HAVE_ASYNC_ST 1
#endif
#endif

#define SA 132   // LDS row stride (floats) for expA rows   (even: b64-aligned)
#define SBT 130  // LDS row stride (floats) for B' columns  (even: b64-aligned)

__global__ __launch_bounds__(256) void goom_lme_kernel(
    const float* __restrict__ Af,   // complex64 viewed as float pairs
    const float* __restrict__ Bf,   // complex64 viewed as float pairs
    float* __restrict__ Of) {       // complex64 viewed as float pairs
  // LDS: sA (128*SA) | sBt (128*SBT, transposed B') | sRed(512) | sAmax(128) | sBmax(64)
  // sOut (16384 floats) overlays sA after the GEMM (sA dead by then).
  __shared__ __align__(16) float smem[128 * SA + 128 * SBT + 512 + 128 + 64];
  float* sA    = smem;
  float* sBt   = smem + 128 * SA;     // sBt[col*SBT + row]
  float* sRed  = sBt + 128 * SBT;
  float* sAmax = sRed + 512;
  float* sBmax = sAmax + 128;
  float* sOut  = smem;  // overlay

  const int tid = threadIdx.x;
  const int w   = tid >> 5;   // wave id 0..7
  const int l   = tid & 31;   // lane id
  const size_t tileOff = (size_t)blockIdx.x * 16384;  // floats per [128,64] c64 tile

  // ---------------- Phase A: load A tile, row maxima, build expA ------------
  // float4 index t+256*i covers row m = w + 8*i, complex cols k0 = 2*l, 2*l+1.
  {
    float4 ra[16];
    const float4* A4 = (const float4*)(Af + tileOff);
#pragma unroll
    for (int i = 0; i < 16; ++i) ra[i] = A4[tid + 256 * i];

#pragma unroll
    for (int i = 0; i < 16; ++i) {
      const int m = w + 8 * i;
      float pm = fmaxf(ra[i].x, ra[i].z);  // real parts
#pragma unroll
      for (int s = 16; s > 0; s >>= 1) pm = fmaxf(pm, __shfl_xor(pm, s, 32));
      if (m >= 64) pm = fmaxf(pm, 0.0f);       // identity block contributes log(1)=0
      if (!(pm >= -3.0e38f)) pm = 0.0f;        // -inf / NaN guard (jnp.where(isfinite))
      if (l == 0) sAmax[m] = pm;
      const int k0 = 2 * l;
      const float e0 = __expf(ra[i].x - pm);
      const float e1 = __expf(ra[i].z - pm);
      // expA packed as [Ar | Ai], 128 cols, row-major
      sA[m * SA + k0]          = e0 * __cosf(ra[i].y);
      sA[m * SA + k0 + 1]      = e1 * __cosf(ra[i].w);
      sA[m * SA + 64 + k0]     = e0 * __sinf(ra[i].y);
      sA[m * SA + 64 + k0 + 1] = e1 * __sinf(ra[i].w);
    }
  }

  // ---------------- Phase B: load B tile, col maxima, build packed B' -------
  {
    float4 rb[16];
    const float4* B4 = (const float4*)(Bf + tileOff);
#pragma unroll
    for (int i = 0; i < 16; ++i) rb[i] = B4[tid + 256 * i];

    // per-wave partial column maxima (lane l owns cols 2l, 2l+1)
    float cm0 = -__builtin_inff(), cm1 = -__builtin_inff();
#pragma unroll
    for (int i = 0; i < 16; ++i) {
      cm0 = fmaxf(cm0, rb[i].x);
      cm1 = fmaxf(cm1, rb[i].z);
    }
    sRed[w * 64 + 2 * l]     = cm0;
    sRed[w * 64 + 2 * l + 1] = cm1;
    __syncthreads();
    if (tid < 64) {
      float bm = sRed[tid];
#pragma unroll
      for (int w2 = 1; w2 < 8; ++w2) bm = fmaxf(bm, sRed[w2 * 64 + tid]);
      if (!(bm >= -3.0e38f)) bm = 0.0f;
      sBmax[tid] = bm;
    }
    __syncthreads();

    // packed B' = [[Br, Bi], [-Bi, Br]] from top 64 rows of B (i < 8),
    // stored TRANSPOSED: sBt[col*SBT + row]
    const int j0 = 2 * l;
    const float bm0 = sBmax[j0], bm1 = sBmax[j0 + 1];
#pragma unroll
    for (int i = 0; i < 8; ++i) {
      const int k = w + 8 * i;  // < 64
      const float e0  = __expf(rb[i].x - bm0);
      const float e1  = __expf(rb[i].z - bm1);
      const float br0 = e0 * __cosf(rb[i].y), bi0 = e0 * __sinf(rb[i].y);
      const float br1 = e1 * __cosf(rb[i].w), bi1 = e1 * __sinf(rb[i].w);
      // B'[k][j] = br ; B'[k][64+j] = bi ; B'[64+k][j] = -bi ; B'[64+k][64+j] = br
      sBt[j0 * SBT + k]              = br0;
      sBt[(j0 + 1) * SBT + k]        = br1;
      sBt[(64 + j0) * SBT + k]       = bi0;
      sBt[(64 + j0 + 1) * SBT + k]   = bi1;
      sBt[j0 * SBT + 64 + k]         = -bi0;
      sBt[(j0 + 1) * SBT + 64 + k]   = -bi1;
      sBt[(64 + j0) * SBT + 64 + k]      = br0;
      sBt[(64 + j0 + 1) * SBT + 64 + k]  = br1;
    }
  }
  __syncthreads();

  // ---------------- Phase C: WMMA GEMM  C(128x128) = expA @ B' --------------
  // Wave w owns output rows [16w,16w+16), all 8 column tiles.
  // A frag 16x4 f32: lane l -> row 16w+(l&15); VGPR r holds K = (l>>4)*2 + r.
  // B frag 4x16 f32: lane l -> col 16ct+(l&15); VGPR r holds K = (l>>4)*2 + r.
  // Both fragments are contiguous b64 loads from LDS.
  const int rowA = 16 * w + (l & 15);
  const int koff = (l >> 4) * 2;
  const int cb   = l & 15;
  v8f acc[8] = {};
  for (int kb = 0; kb < 128; kb += 4) {
    const v2f a = *(const v2f*)&sA[rowA * SA + kb + koff];
#pragma unroll
    for (int ct = 0; ct < 8; ++ct) {
      const v2f b = *(const v2f*)&sBt[(16 * ct + cb) * SBT + kb + koff];
      acc[ct] = __builtin_amdgcn_wmma_f32_16x16x4_f32(
          /*neg_a=*/false, a, /*neg_b=*/false, b,
          /*c_mod=*/(short)0, acc[ct], /*reuse_a=*/false, /*reuse_b=*/false);
    }
  }
  __syncthreads();  // sA dead -> sOut overlay becomes safe

  // ---------------- Phase D: epilogue (identity correction + log) -----------
  // C cols 0..63 = Re, 64..127 = Im. C/D layout: lanes 0-15 hold M=r, lanes
  // 16-31 hold M=r+8 in VGPR r; N = 16ct + (l&15).
  {
    const float2* Bc = (const float2*)(Bf + tileOff);
#pragma unroll
    for (int ct = 0; ct < 4; ++ct) {
#pragma unroll
      for (int r = 0; r < 8; ++r) {
        const int m = 16 * w + ((l < 16) ? r : r + 8);
        const int j = 16 * ct + cb;
        float pre = acc[ct][r];
        float pim = acc[ct + 4][r];
        const float am = sAmax[m], bm = sBmax[j];
        if (m >= 64) {
          // identity-row term: exp(-a_max[m]) * exp(B[m][j] - b_max[j])
          const float2 braw = Bc[m * 64 + j];  // hits L2 (just streamed)
          const float e = __expf(braw.x - bm - am);
          pre += e * __cosf(braw.y);
          pim += e * __sinf(braw.y);
        }
        sOut[(m * 64 + j) * 2]     = __logf(hypotf(pre, pim)) + am + bm;
        sOut[(m * 64 + j) * 2 + 1] = atan2f(pim, pre);
      }
    }
  }
  __syncthreads();

  // ---------------- Phase E: stream results LDS -> HBM ----------------------
  float* outT = Of + tileOff;
#if defined(HAVE_ASYNC_ST)
#pragma unroll
  for (int i = 0; i < 16; ++i) {
    const int f4 = tid + 256 * i;
    __builtin_amdgcn_global_store_async_from_lds_b128(
        (AS1 v4i*)(uint64_t)(outT + 4 * f4),
        (AS3 v4i*)(uint32_t)(uint64_t)(sOut + 4 * f4),
        0, 0);
  }
#if __has_builtin(__builtin_amdgcn_s_wait_asynccnt)
  __builtin_amdgcn_s_wait_asynccnt(0);
#else
  asm volatile("s_wait_asynccnt 0" ::: "memory");
#endif
#else
  float4* O4 = (float4*)outT;
#pragma unroll
  for (int i = 0; i < 16; ++i) O4[tid + 256 * i] = ((const float4*)sOut)[tid + 256 * i];
#endif
}

extern "C" void kernel_launch(void* const* d_in, const int* in_sizes, int n_in,
                              void* d_out, int out_size, void* d_ws, size_t ws_size,
                              hipStream_t stream) {
  (void)in_sizes; (void)n_in; (void)d_ws; (void)ws_size;
  const float* A = (const float*)d_in[0];  // log_A1_atop_B1 (complex64 pairs)
  const float* B = (const float*)d_in[1];  // log_A2_atop_B2 (complex64 pairs)
  float* O = (float*)d_out;
  // out_size = floats in output = batch * 128 * 64 * 2
  int batch = out_size / 16384;
  if (batch <= 0) batch = 2048;
  goom_lme_kernel<<<dim3(batch), dim3(256), 0, stream>>>(A, B, O);
}